// ElasticTransform_4475355922850
// MI455X (gfx1250) — compile-verified
//
#include <hip/hip_runtime.h>
#include <math.h>

// ---------------------------------------------------------------------------
// ElasticTransform on MI455X (gfx1250, wave32)
//   pass 1: horizontal 63-tap Gaussian as banded GEMM via V_WMMA_F32_16X16X4_F32
//   pass 2: vertical   63-tap Gaussian as banded GEMM via V_WMMA_F32_16X16X4_F32
//   pass 3: bilinear grid_sample (gather, VALU-bound)
// Branchless operand fetch (clamped addresses + cndmask zeroing, guard-banded
// LDS weight table). All global accesses use uniform SGPR base + unsigned
// 32-bit lane offsets so the backend can pick saddr (GVS) addressing instead
// of 64-bit VGPR address pairs.
// ---------------------------------------------------------------------------

typedef __attribute__((ext_vector_type(2))) float v2f;
typedef __attribute__((ext_vector_type(8))) float v8f;

#define BATCH   8
#define CIMG    3
#define CNOISE  2
#define HH      512
#define WW      512
#define KS      63
#define RAD     31
#define SIG     32.0f
#define NCHUNK  20      // K padded to 80 = 20 * 4
#define GPAD    16      // guard band on each side of the weight table

// Guard-banded unnormalized Gaussian band table in LDS:
//   gt[i] = exp(-((i-GPAD)-31)^2 / (2*sig^2)) for (i-GPAD) in [0,63), else 0
//   gt[96] = 1/sum  (normalization folded in at store time)
__device__ __forceinline__ float init_gauss(float* gt) {
    unsigned tid = threadIdx.x;
    if (tid < 96u) {
        int d = (int)tid - GPAD;
        float fd = (float)(d - RAD);
        float e = expf(-(fd * fd) / (2.0f * SIG * SIG));
        gt[tid] = ((unsigned)d < (unsigned)KS) ? e : 0.0f;
    }
    __syncthreads();
    if (tid == 0u) {
        float s = 0.0f;
        for (int i = 0; i < KS; ++i) s += gt[GPAD + i];
        gt[96] = 1.0f / s;
    }
    __syncthreads();
    return gt[96];
}

__device__ __forceinline__ unsigned uclamp(int v, int lo, int hi) {
    return (unsigned)min(max(v, lo), hi);
}

// ---------------------------------------------------------------------------
// Pass 1: horizontal blur of (noise*2-1), zero pad.
// One wave = one 16x16 tile: O = A(16x80 input window) x B(80x16 band matrix)
// ---------------------------------------------------------------------------
__global__ void __launch_bounds__(256)
et_hblur_wmma(const float* __restrict__ noise, float* __restrict__ tmp) {
    __shared__ float gt[104];
    float invs = init_gauss(gt);

    unsigned tid  = threadIdx.x;
    unsigned lane = tid & 31u;
    unsigned wave = tid >> 5;
    unsigned tile = blockIdx.x * (blockDim.x >> 5) + wave;   // 16384 tiles

    unsigned tx = tile & 31u;
    unsigned ty = (tile >> 5) & 31u;
    unsigned bc = tile >> 10;                                // b*2 + ch

    unsigned half = lane >> 4;     // 0: K pair {0,1}, 1: K pair {2,3}
    unsigned q    = lane & 15u;    // A: row m ; B/C: col n
    int      xbase = (int)(tx * 16u) - RAD;
    unsigned ybase = ty * 16u;

    // uniform-base + 32-bit offset addressing
    unsigned rowoff = bc * (HH * WW) + (ybase + q) * WW;

    v2f afrag[NCHUNK], bfrag[NCHUNK];
    #pragma unroll
    for (int k = 0; k < NCHUNK; ++k) {
        int j0 = 4 * k + (int)(half * 2u);
        // A: clamped unconditional loads, cndmask zero pad, fold 2n-1
        int c0 = xbase + j0;
        int c1 = c0 + 1;
        float r0 = noise[rowoff + uclamp(c0, 0, WW - 1)];
        float r1 = noise[rowoff + uclamp(c1, 0, WW - 1)];
        float a0 = ((unsigned)c0 < (unsigned)WW) ? fmaf(2.0f, r0, -1.0f) : 0.0f;
        float a1 = ((unsigned)c1 < (unsigned)WW) ? fmaf(2.0f, r1, -1.0f) : 0.0f;
        v2f a = {a0, a1};
        afrag[k] = a;
        // B: band matrix W[j, n] = g[j - n]; index always in [1,95]
        unsigned d0 = (unsigned)(j0 - (int)q + GPAD);
        v2f bm = {gt[d0], gt[d0 + 1u]};
        bfrag[k] = bm;
    }

    v8f acc = {};
    #pragma unroll
    for (int k = 0; k < NCHUNK; ++k) {
        acc = __builtin_amdgcn_wmma_f32_16x16x4_f32(
            false, afrag[k], false, bfrag[k], (short)0, acc, false, false);
    }

    // C layout: VGPR i -> M = i + half*8, N = q
    unsigned dstoff = bc * (HH * WW) + ybase * WW + tx * 16u + q + half * (8u * WW);
    #pragma unroll
    for (int i = 0; i < 8; ++i) {
        tmp[dstoff + (unsigned)i * WW] = acc[i] * invs;
    }
}

// ---------------------------------------------------------------------------
// Pass 2: vertical blur of tmp -> disp stored as [B,H,W,2].
// O = A(16x80 band weights) x B(80x16 column window of tmp)
// ---------------------------------------------------------------------------
__global__ void __launch_bounds__(256)
et_vblur_wmma(const float* __restrict__ tmp, float* __restrict__ disp) {
    __shared__ float gt[104];
    float invs = init_gauss(gt);

    unsigned tid  = threadIdx.x;
    unsigned lane = tid & 31u;
    unsigned wave = tid >> 5;
    unsigned tile = blockIdx.x * (blockDim.x >> 5) + wave;

    unsigned tx = tile & 31u;
    unsigned ty = (tile >> 5) & 31u;
    unsigned bc = tile >> 10;
    unsigned ch = bc & 1u;
    unsigned b  = bc >> 1;

    unsigned half = lane >> 4;
    unsigned q    = lane & 15u;
    int      ybase = (int)(ty * 16u) - RAD;
    unsigned x0    = tx * 16u;

    unsigned coloff = bc * (HH * WW) + x0 + q;   // + row*WW added per element

    v2f afrag[NCHUNK], bfrag[NCHUNK];
    #pragma unroll
    for (int k = 0; k < NCHUNK; ++k) {
        int j0 = 4 * k + (int)(half * 2u);
        // A: band weights A[m=q, j] = g[j - q]; index always in [1,95]
        unsigned d0 = (unsigned)(j0 - (int)q + GPAD);
        v2f a = {gt[d0], gt[d0 + 1u]};
        afrag[k] = a;
        // B: tmp rows j0, j0+1 at column x0+q; clamp row, cndmask zero pad
        int r0 = ybase + j0;
        int r1 = r0 + 1;
        float v0 = tmp[coloff + uclamp(r0, 0, HH - 1) * (unsigned)WW];
        float v1 = tmp[coloff + uclamp(r1, 0, HH - 1) * (unsigned)WW];
        float b0 = ((unsigned)r0 < (unsigned)HH) ? v0 : 0.0f;
        float b1 = ((unsigned)r1 < (unsigned)HH) ? v1 : 0.0f;
        v2f bm = {b0, b1};
        bfrag[k] = bm;
    }

    v8f acc = {};
    #pragma unroll
    for (int k = 0; k < NCHUNK; ++k) {
        acc = __builtin_amdgcn_wmma_f32_16x16x4_f32(
            false, afrag[k], false, bfrag[k], (short)0, acc, false, false);
    }

    // store interleaved [B,H,W,2]; C layout: VGPR i -> y = ty*16 + i + half*8
    unsigned dstoff = ((b * HH + ty * 16u + half * 8u) * WW + x0 + q) * 2u + ch;
    #pragma unroll
    for (int i = 0; i < 8; ++i) {
        disp[dstoff + (unsigned)i * (2u * WW)] = acc[i] * invs;
    }
}

// ---------------------------------------------------------------------------
// Pass 3: bilinear grid_sample (align_corners=False, zeros padding).
// One thread per (b,y,x), 3 channels each.
// ---------------------------------------------------------------------------
__global__ void __launch_bounds__(256)
et_warp(const float* __restrict__ image, const float* __restrict__ disp,
        float* __restrict__ out) {
    unsigned idx = blockIdx.x * blockDim.x + threadIdx.x;
    if (idx >= (unsigned)(BATCH * HH * WW)) return;
    unsigned x = idx & (WW - 1u);
    unsigned y = (idx >> 9) & (HH - 1u);
    unsigned b = idx >> 18;

    v2f d2 = *(const v2f*)(disp + (size_t)idx * 2u);   // global_load_b64
    float dx = d2.x, dy = d2.y;

    float bx = -1.0f + 2.0f * (float)x / (float)(WW - 1);
    float by = -1.0f + 2.0f * (float)y / (float)(HH - 1);
    float gx = fminf(fmaxf(bx + dx, -1.0f), 1.0f);
    float gy = fminf(fmaxf(by + dy, -1.0f), 1.0f);

    float xf = ((gx + 1.0f) * (float)WW - 1.0f) * 0.5f;
    float yf = ((gy + 1.0f) * (float)HH - 1.0f) * 0.5f;
    float xfl = floorf(xf), yfl = floorf(yf);
    int ix0 = (int)xfl, iy0 = (int)yfl;
    int ix1 = ix0 + 1,  iy1 = iy0 + 1;
    float wx1 = xf - xfl, wx0 = 1.0f - wx1;
    float wy1 = yf - yfl, wy0 = 1.0f - wy1;

    bool vx0 = (ix0 >= 0 && ix0 < WW), vx1 = (ix1 >= 0 && ix1 < WW);
    bool vy0 = (iy0 >= 0 && iy0 < HH), vy1 = (iy1 >= 0 && iy1 < HH);
    unsigned cx0 = uclamp(ix0, 0, WW - 1), cx1 = uclamp(ix1, 0, WW - 1);
    unsigned cy0 = uclamp(iy0, 0, HH - 1), cy1 = uclamp(iy1, 0, HH - 1);

    float w00 = wy0 * wx0 * (float)(vy0 && vx0);
    float w01 = wy0 * wx1 * (float)(vy0 && vx1);
    float w10 = wy1 * wx0 * (float)(vy1 && vx0);
    float w11 = wy1 * wx1 * (float)(vy1 && vx1);

    unsigned imbase  = b * (CIMG * HH * WW);
    unsigned outbase = imbase + y * WW + x;
    #pragma unroll
    for (unsigned c = 0; c < CIMG; ++c) {
        unsigned imc = imbase + c * (HH * WW);
        float v = w00 * image[imc + cy0 * WW + cx0]
                + w01 * image[imc + cy0 * WW + cx1]
                + w10 * image[imc + cy1 * WW + cx0]
                + w11 * image[imc + cy1 * WW + cx1];
        out[outbase + c * (HH * WW)] = v;
    }
}

extern "C" void kernel_launch(void* const* d_in, const int* in_sizes, int n_in,
                              void* d_out, int out_size, void* d_ws, size_t ws_size,
                              hipStream_t stream) {
    const float* image = (const float*)d_in[0];   // [8,3,512,512]
    const float* noise = (const float*)d_in[1];   // [8,2,512,512]
    float* out    = (float*)d_out;
    float* warped = out;                                        // 8*3*512*512
    float* disp   = out + (size_t)BATCH * CIMG * HH * WW;       // 8*512*512*2
    float* tmp    = (float*)d_ws;                               // 8*2*512*512 f32

    const int tiles  = BATCH * CNOISE * (HH / 16) * (WW / 16);  // 16384
    const int wpb    = 8;                                       // waves / block
    const int blocks = tiles / wpb;                             // 2048

    et_hblur_wmma<<<blocks, 32 * wpb, 0, stream>>>(noise, tmp);
    et_vblur_wmma<<<blocks, 32 * wpb, 0, stream>>>(tmp, disp);

    const int n = BATCH * HH * WW;
    et_warp<<<(n + 255) / 256, 256, 0, stream>>>(image, disp, warped);
}